// rotary_13872744366393
// MI455X (gfx1250) — compile-verified
//
#include <hip/hip_runtime.h>
#include <math.h>

// Problem constants (from reference)
#define DIMS     1024
#define HEADS    16
#define HDIM     64
#define ROT      32
#define BATCH    4
#define SEQ      8192

typedef __attribute__((ext_vector_type(2))) float v2f;
typedef __attribute__((ext_vector_type(8))) float v8f;

// -------------------------------------------------------------------------
// Kernel 1: M = G_comb(thetas * tscale, pairs) @ matrix   (64x64 -> d_ws)
// acc starts as identity; acc @ G (Givens) only mixes columns i and j of
// each row, so one thread per row needs no synchronization at all.
// -------------------------------------------------------------------------
__global__ void __launch_bounds__(64)
build_M_kernel(const float* __restrict__ matrix,
               const float* __restrict__ thetas,
               const float* __restrict__ tscale,
               const int*   __restrict__ pairs,
               float*       __restrict__ Mout) {
    __shared__ float acc[HDIM][HDIM];
    const int t = threadIdx.x;              // row 0..63
    for (int c = 0; c < HDIM; ++c) acc[t][c] = (t == c) ? 1.0f : 0.0f;

    const float ts = tscale[0];
    for (int g = 0; g < ROT; ++g) {
        const int   i  = pairs[2 * g];
        const int   j  = pairs[2 * g + 1];
        const float th = thetas[g] * ts;
        const float cs = cosf(th), sn = sinf(th);
        if (i != j) {
            const float vi = acc[t][i], vj = acc[t][j];
            acc[t][i] =  cs * vi + sn * vj;
            acc[t][j] = -sn * vi + cs * vj;
        } else {
            // reference's .at chain degenerates: G[i,i] ends up = sin(th)
            acc[t][i] *= sn;
        }
    }
    // M[t][n] = sum_m acc[t][m] * matrix[m][n]
    for (int n = 0; n < HDIM; ++n) {
        float s = 0.0f;
        for (int m = 0; m < HDIM; ++m)
            s = fmaf(acc[t][m], matrix[m * HDIM + n], s);
        Mout[t * HDIM + n] = s;
    }
}

// -------------------------------------------------------------------------
// Kernel 2: fused  (x-tile @ M)  ->  RoPE  ->  * sqrt(DIMS)  -> out
// 256 threads = 8 waves; each wave computes one 16-row tile = one (b,s)
// position with all 16 heads, via v_wmma_f32_16x16x4_f32.
//
// M is pre-swizzled into WMMA B-fragment layout in LDS so every B fragment
// is ONE bank-conflict-free ds_load_b64 into an even-aligned VGPR pair
// (no register repacking movs in the hot loop).
// -------------------------------------------------------------------------
#define STG 66   // LDS row stride for C staging (bank-conflict padding)

__global__ void __launch_bounds__(256)
rope_gemm_kernel(const float* __restrict__ x,
                 const float* __restrict__ Mg,
                 const float* __restrict__ invf,
                 float*       __restrict__ out) {
    // Bl[(kk*4 + t)*64 + lane*2 + v] = M[4kk + 2*(lane>>4) + v][16t + lane&15]
    __shared__ __align__(16) float Bl[16 * 4 * 64];          // 16 KB
    __shared__ __align__(16) float stg[8][16 * STG];         // 33 KB
    __shared__ __align__(16) float sctab[8][64];             // [0..31]=cos [32..63]=sin

    const int tid  = threadIdx.x;
    const int wave = tid >> 5;
    const int lane = tid & 31;

    // Cooperative pre-swizzle of M into B-fragment layout (once per block;
    // all blocks read the same 16 KB -> lives in the 192 MB L2).
    for (int idx = tid; idx < 16 * 4 * 64; idx += 256) {
        const int v  = idx & 1;
        const int ln = (idx >> 1) & 31;
        const int t  = (idx >> 6) & 3;
        const int kk = idx >> 8;
        const int krow = 4 * kk + ((ln >> 4) << 1) + v;
        const int col  = 16 * t + (ln & 15);
        Bl[idx] = Mg[krow * HDIM + col];
    }
    __syncthreads();

    const int  tile    = blockIdx.x * 8 + wave;     // one (b,s) pair
    const int  s       = tile & (SEQ - 1);
    const long rowbase = (long)tile * 16;           // 16 head-rows of 64

    // Per-tile sin/cos table (single position s for the whole tile)
    if (lane < ROT) {
        const float a = (float)s * invf[lane];
        sctab[wave][lane]      = cosf(a);
        sctab[wave][lane + 32] = sinf(a);
    }

    // A fragments: ISA 16x4 fp32 A layout.
    // lane L holds row (L%16), K = 4*kk + 2*(L/16) + {0,1}  -> b64 loads
    const float* xrow = x + (rowbase + (lane & 15)) * HDIM + ((lane >> 4) << 1);
    v2f afrag[16];
#pragma unroll
    for (int kk = 0; kk < 16; ++kk)
        afrag[kk] = *(const v2f*)(xrow + kk * 4);

    v8f c0 = {}, c1 = {}, c2 = {}, c3 = {};
    const float* Bp = Bl + lane * 2;
#pragma unroll
    for (int kk = 0; kk < 16; ++kk) {
        const v2f b0 = *(const v2f*)(Bp + (kk * 4 + 0) * 64);
        const v2f b1 = *(const v2f*)(Bp + (kk * 4 + 1) * 64);
        const v2f b2 = *(const v2f*)(Bp + (kk * 4 + 2) * 64);
        const v2f b3 = *(const v2f*)(Bp + (kk * 4 + 3) * 64);
        c0 = __builtin_amdgcn_wmma_f32_16x16x4_f32(false, afrag[kk], false, b0,
                                                   (short)0, c0, false, false);
        c1 = __builtin_amdgcn_wmma_f32_16x16x4_f32(false, afrag[kk], false, b1,
                                                   (short)0, c1, false, false);
        c2 = __builtin_amdgcn_wmma_f32_16x16x4_f32(false, afrag[kk], false, b2,
                                                   (short)0, c2, false, false);
        c3 = __builtin_amdgcn_wmma_f32_16x16x4_f32(false, afrag[kk], false, b3,
                                                   (short)0, c3, false, false);
    }

    // Stage C tiles to LDS: VGPR r of lane L is out[r + 8*(L/16)][16t + L%16]
    {
        const int rbase = (lane >> 4) << 3;
        const int col   = lane & 15;
        float*    st    = stg[wave];
#pragma unroll
        for (int r = 0; r < 8; ++r) {
            st[(rbase + r) * STG + col]      = c0[r];
            st[(rbase + r) * STG + col + 16] = c1[r];
            st[(rbase + r) * STG + col + 32] = c2[r];
            st[(rbase + r) * STG + col + 48] = c3[r];
        }
    }
    __syncthreads();

    // RoPE epilogue: out[..., n]    = y[2n]*cos - y[2n+1]*sin      (n < 32)
    //                out[..., n+32] = y[2n]*sin + y[2n+1]*cos
    // 1024 outputs per wave, 32 per lane, coalesced 128B stores.
    const float* st = stg[wave];
    const float* sc = sctab[wave];
    float*       op = out + rowbase * HDIM;
#pragma unroll
    for (int i = 0; i < 32; ++i) {
        const int flat = i * 32 + lane;
        const int row  = flat >> 6;
        const int col  = flat & 63;
        const int n    = col & 31;
        const v2f xy = *(const v2f*)(st + row * STG + 2 * n);
        const float cs = sc[n], sn = sc[n + 32];
        const float val = (col < 32) ? (xy.x * cs - xy.y * sn)
                                     : (xy.x * sn + xy.y * cs);
        op[flat] = val * 32.0f;   // sqrt(DIMS) = sqrt(1024)
    }
}

// -------------------------------------------------------------------------
extern "C" void kernel_launch(void* const* d_in, const int* in_sizes, int n_in,
                              void* d_out, int out_size, void* d_ws, size_t ws_size,
                              hipStream_t stream) {
    const float* x      = (const float*)d_in[0];
    const float* matrix = (const float*)d_in[1];
    const float* thetas = (const float*)d_in[2];
    const float* tscale = (const float*)d_in[3];
    const float* invf   = (const float*)d_in[4];
    const int*   pairs  = (const int*)d_in[5];
    float*       Mws    = (float*)d_ws;            // 64*64 fp32 = 16 KB

    build_M_kernel<<<1, 64, 0, stream>>>(matrix, thetas, tscale, pairs, Mws);

    const int tiles  = BATCH * SEQ;                // 32768 (16 rows each)
    const int blocks = tiles / 8;                  // 8 waves/block
    rope_gemm_kernel<<<blocks, 256, 0, stream>>>(x, Mws, invf, (float*)d_out);
}